// ConcentrationLoss_84945863180964
// MI455X (gfx1250) — compile-verified
//
#include <hip/hip_runtime.h>
#include <hip/hip_bf16.h>

typedef __attribute__((ext_vector_type(2))) float v2f;
typedef __attribute__((ext_vector_type(8))) float v8f;

#define HW   4096   // 64*64
#define WDIM 64
#define WIN  8
#define STRIDE 4
#define OH   15     // (64-8)/4+1
#define NPATCH (OH*OH)  // 225

// ---------------------------------------------------------------------------
// Kernel 1: coord1[b, m, n] = sum_k g[b,m,k] * aff[b,k,n],  m in {0,1}
//   g[b,0,k] = grid[b,k,0], g[b,1,k] = grid[b,k,1]   (grid flat: (b, hw, 2))
// One wave computes one 16-column tile via V_WMMA_F32_16X16X4_F32:
//   A (16x4 f32): lanes 0-15 -> (M=lane, K=k0+0 / k0+1), lanes 16-31 -> (M, K=k0+2 / k0+3)
//   B (4x16 f32): lanes 0-15 -> (K=k0+0/1, N=lane), lanes 16-31 -> (K=k0+2/3, N=lane-16)
//   D (16x16):    VGPR r, lanes 0-15 -> (M=r, N=lane)  => rows 0,1 live in c[0], c[1]
// ---------------------------------------------------------------------------
__global__ __launch_bounds__(256) void coord_gemm_wmma(
    const float* __restrict__ aff, const float* __restrict__ grid,
    float* __restrict__ coord1)
{
  __shared__ float g_lds[2 * HW];   // 32 KB: interleaved (k, comp) copy of grid[b]
  const int b   = blockIdx.y;
  const int tid = threadIdx.x;

  const float* gsrc = grid + (size_t)b * (2 * HW);
  for (int i = tid; i < 2 * HW; i += 256) g_lds[i] = gsrc[i];
  __syncthreads();

  const int lane = tid & 31;
  const int wave = tid >> 5;
  const int m    = lane & 15;           // M index (A) / N offset (B)
  const int koff = (lane >> 4) << 1;    // 0 for lanes 0-15, 2 for lanes 16-31
  const int n0   = blockIdx.x * 128 + wave * 16;
  const int n    = n0 + m;
  const bool am_live = (m < 2);

  const float* __restrict__ affb = aff + (size_t)b * HW * HW;

  v8f c = {0.f, 0.f, 0.f, 0.f, 0.f, 0.f, 0.f, 0.f};

  #pragma unroll 4
  for (int k0 = 0; k0 < HW; k0 += 4) {
    const int ka = k0 + koff;
    v2f bm, am;
    bm.x = affb[(size_t)ka * HW + n];          // B vgpr0: row ka
    bm.y = affb[(size_t)(ka + 1) * HW + n];    // B vgpr1: row ka+1
    const int gb = ka * 2;
    float a0 = g_lds[am_live ? (gb + m)     : 0];
    float a1 = g_lds[am_live ? (gb + 2 + m) : 0];
    am.x = am_live ? a0 : 0.f;
    am.y = am_live ? a1 : 0.f;
    c = __builtin_amdgcn_wmma_f32_16x16x4_f32(
        /*neg_a=*/false, am, /*neg_b=*/false, bm,
        /*c_mod=*/(short)0, c, /*reuse_a=*/false, /*reuse_b=*/false);
  }

  if (lane < 16) {
    coord1[(size_t)b * (2 * HW) + n0 + lane]      = c[0];  // m=0 row
    coord1[(size_t)b * (2 * HW) + HW + n0 + lane] = c[1];  // m=1 row
  }
}

// ---------------------------------------------------------------------------
// Kernel 2: per-(b,c) slice: sum over 8x8/stride-4 windows of sum((x-mean)^2)
// ---------------------------------------------------------------------------
__global__ __launch_bounds__(256) void loss_partial(
    const float* __restrict__ coord1, float* __restrict__ partial)
{
  __shared__ float s[HW];      // 16 KB slice
  __shared__ float red[256];
  const int tid = threadIdx.x;
  const float* src = coord1 + (size_t)blockIdx.x * HW;
  for (int i = tid; i < HW; i += 256) s[i] = src[i];
  __syncthreads();

  float acc = 0.f;
  for (int p = tid; p < NPATCH; p += 256) {
    const int pi = p / OH, pj = p % OH;
    const int r0 = pi * STRIDE, c0 = pj * STRIDE;
    float sum = 0.f;
    #pragma unroll
    for (int r = 0; r < WIN; ++r)
      #pragma unroll
      for (int cc = 0; cc < WIN; ++cc)
        sum += s[(r0 + r) * WDIM + (c0 + cc)];
    const float mean = sum * (1.0f / (WIN * WIN));
    float sq = 0.f;
    #pragma unroll
    for (int r = 0; r < WIN; ++r)
      #pragma unroll
      for (int cc = 0; cc < WIN; ++cc) {
        const float d = s[(r0 + r) * WDIM + (c0 + cc)] - mean;
        sq += d * d;
      }
    acc += sq;
  }

  red[tid] = acc;
  __syncthreads();
  for (int off = 128; off > 0; off >>= 1) {
    if (tid < off) red[tid] += red[tid + off];
    __syncthreads();
  }
  if (tid == 0) partial[blockIdx.x] = red[0];
}

// ---------------------------------------------------------------------------
// Kernel 3: final scalar = sum(partials) / (b * c * 225 * 64)
// ---------------------------------------------------------------------------
__global__ void finalize(const float* __restrict__ partial,
                         float* __restrict__ out, int nslices)
{
  if (threadIdx.x == 0) {
    float t = 0.f;
    for (int i = 0; i < nslices; ++i) t += partial[i];
    out[0] = t / ((float)nslices * NPATCH * WIN * WIN);
  }
}

extern "C" void kernel_launch(void* const* d_in, const int* in_sizes, int n_in,
                              void* d_out, int out_size, void* d_ws, size_t ws_size,
                              hipStream_t stream) {
  const float* aff  = (const float*)d_in[0];   // (b, 4096, 4096) f32
  const float* grid = (const float*)d_in[1];   // (b, 64, 64, 2)  f32
  const int b = in_sizes[1] / (2 * HW);        // = 8

  float* coord1  = (float*)d_ws;               // b*2*4096 floats = 256 KB
  float* partial = coord1 + (size_t)b * 2 * HW;
  float* out     = (float*)d_out;

  dim3 g1(HW / 128, b);                        // 32 col-blocks x 8 batches
  coord_gemm_wmma<<<g1, 256, 0, stream>>>(aff, grid, coord1);

  loss_partial<<<dim3(b * 2), 256, 0, stream>>>(coord1, partial);

  finalize<<<1, 32, 0, stream>>>(partial, out, b * 2);
}